// Model_59768764891685
// MI455X (gfx1250) — compile-verified
//
#include <hip/hip_runtime.h>
#include <hip/hip_bf16.h>
#include <math.h>

typedef __attribute__((ext_vector_type(2))) float v2f;
typedef __attribute__((ext_vector_type(4))) float v4f;
typedef __attribute__((ext_vector_type(8))) float v8f;
typedef __attribute__((ext_vector_type(4))) unsigned int v4u;

#define N_NODES   8192
#define N_EDGES   262144
#define HID       128
#define OUTD      16

// ---------------------------------------------------------------------------
// Repack W[K,Nc] (row-major) into WMMA B-fragment order:
//   Wpk[(kstep*tilesN + tN)*32 + lane] = { W[4*kstep+kb][col], W[4*kstep+kb+1][col] }
//   kb = 2*(lane>>4), col = tN*16 + (lane&15)
// One v2f per thread; coalesced stores; tiny (<= 32K elements per matrix).
// ---------------------------------------------------------------------------
__global__ void __launch_bounds__(256)
pack_w(const float* __restrict__ W, float* __restrict__ Wpk,
       int Nc, int tilesN, int nK4)
{
    const int t = blockIdx.x * 256 + threadIdx.x;
    const int lane = t & 31;
    const int rest = t >> 5;
    const int tN = rest % tilesN;
    const int kstep = rest / tilesN;
    if (kstep >= nK4) return;
    const int kb = (lane >> 4) << 1;
    const int col = (tN << 4) + (lane & 15);
    const int ka = kstep * 4 + kb;
    Wpk[(size_t)t * 2 + 0] = W[(size_t)ka * Nc + col];
    Wpk[(size_t)t * 2 + 1] = W[(size_t)(ka + 1) * Nc + col];
}

// ---------------------------------------------------------------------------
// fp32 WMMA linear on packed weights:
//   out[M,Nc] = cat(in0[M,K0], in1[M,K1]) @ W[K0+K1,Nc] + bias
// One 16x16 tile per wave via V_WMMA_F32_16X16X4_F32. Split K loops (no
// per-iteration concat select); B fragments are single b64 coalesced loads.
// Requires K0 % 4 == 0, M % 16 == 0, Nc % 16 == 0; grids sized exactly so
// EXEC is all-ones at every WMMA.
// ---------------------------------------------------------------------------
__global__ void __launch_bounds__(256)
wmma_linear(const float* __restrict__ in0, int K0,
            const float* __restrict__ in1, int K1,
            const float* __restrict__ Wpk,
            const float* __restrict__ bias,
            float* __restrict__ out, int Nc, int tilesN)
{
    const int lane = threadIdx.x & 31;
    const int wave = threadIdx.x >> 5;
    const int tile = blockIdx.x * 8 + wave;
    const int tM = tile / tilesN;
    const int tN = tile - tM * tilesN;
    const int mrow = (tM << 4) + (lane & 15);
    const int col  = (tN << 4) + (lane & 15);
    const int kb   = (lane >> 4) << 1;          // 0 for lanes 0-15, 2 for 16-31

    const float* __restrict__ a0 = in0 + (size_t)mrow * K0 + kb;
    const v2f*   __restrict__ bp = (const v2f*)Wpk + (size_t)tN * 32 + lane;
    const int bstride = tilesN * 32;            // v2f elements per k-step

    v8f acc = {0.f, 0.f, 0.f, 0.f, 0.f, 0.f, 0.f, 0.f};

    for (int k = 0; k < K0; k += 4) {
        const v2f a = *(const v2f*)(a0 + k);
        const v2f b = *bp;
        bp += bstride;
        acc = __builtin_amdgcn_wmma_f32_16x16x4_f32(false, a, false, b,
                                                    (short)0, acc, false, false);
    }
    if (K1 > 0) {
        const float* __restrict__ a1 = in1 + (size_t)mrow * K1 + kb;
        for (int k = 0; k < K1; k += 4) {
            const v2f a = *(const v2f*)(a1 + k);
            const v2f b = *bp;
            bp += bstride;
            acc = __builtin_amdgcn_wmma_f32_16x16x4_f32(false, a, false, b,
                                                        (short)0, acc, false, false);
        }
    }

    const float bc = bias ? bias[col] : 0.f;
    const int rbase = (tM << 4) + ((lane >> 4) << 3);
#pragma unroll
    for (int r = 0; r < 8; ++r)
        out[(size_t)(rbase + r) * Nc + col] = acc[r] + bc;
}

// ---------------------------------------------------------------------------
// agg init to -inf bit pattern (262144 uint4 = 1048576 floats)
// ---------------------------------------------------------------------------
__global__ void __launch_bounds__(256)
init_agg(v4u* __restrict__ agg)
{
    const int i = blockIdx.x * 256 + threadIdx.x;       // grid = 1024 blocks
    v4u v = {0xFF800000u, 0xFF800000u, 0xFF800000u, 0xFF800000u};
    agg[i] = v;
}

// ---------------------------------------------------------------------------
// Per-edge message + scatter segment-max:
//   msg[e] = A[src[e]] + B[dst[e]] + w[e]*wlast   (bm folded into B)
//   agg[dst] = max(agg[dst], msg) via sign-split native int atomics
// One wave per edge, 4 components/lane. A/B are L2-resident (4 MB each).
// ---------------------------------------------------------------------------
__global__ void __launch_bounds__(256)
edge_msg_max(const float* __restrict__ A, const float* __restrict__ B,
             const float* __restrict__ wlast,
             const int* __restrict__ src, const int* __restrict__ dst,
             const float* __restrict__ w, float* __restrict__ agg)
{
    const int e = blockIdx.x * 8 + (threadIdx.x >> 5);  // grid = E/8 exactly
    const int lane = threadIdx.x & 31;
    const int s = src[e], d = dst[e];
    const float we = w[e];
    const v4f a  = *(const v4f*)(A + (size_t)s * HID + lane * 4);
    const v4f b  = *(const v4f*)(B + (size_t)d * HID + lane * 4);
    const v4f wl = *(const v4f*)(wlast + lane * 4);
    float* p = agg + (size_t)d * HID + lane * 4;
#pragma unroll
    for (int i = 0; i < 4; ++i) {
        const float v = a[i] + b[i] + we * wl[i];
        if (v >= 0.f) atomicMax((int*)(p + i), __float_as_int(v));
        else          atomicMin((unsigned int*)(p + i), __float_as_uint(v));
    }
}

// ---------------------------------------------------------------------------
// agg fixup: +-inf -> 0 (matches reference empty-segment handling)
// ---------------------------------------------------------------------------
__global__ void __launch_bounds__(256)
fixup_agg(float* __restrict__ agg)
{
    const int i = blockIdx.x * 256 + threadIdx.x;       // grid = 4096 blocks
    const float v = agg[i];
    agg[i] = isinf(v) ? 0.f : v;
}

// ---------------------------------------------------------------------------
// Pooled column stats: block c<128 -> new_h col c (mean,max); c>=128 -> y col.
// pooled layout: [mean_h(128) | max_h(128) | mean_y(16) | max_y(16)] = 288
// ---------------------------------------------------------------------------
__global__ void __launch_bounds__(256)
col_reduce(const float* __restrict__ newh, const float* __restrict__ y,
           float* __restrict__ pooled)
{
    __shared__ float ssum[256];
    __shared__ float smax[256];
    const int c = blockIdx.x;                            // grid = 144
    const int tid = threadIdx.x;
    float s = 0.f, m = -INFINITY;
    if (c < HID) {
        for (int r = tid; r < N_NODES; r += 256) {
            const float v = newh[(size_t)r * HID + c];
            s += v; m = fmaxf(m, v);
        }
    } else {
        const int cy = c - HID;
        for (int r = tid; r < N_NODES; r += 256) {
            const float v = y[(size_t)r * OUTD + cy];
            s += v; m = fmaxf(m, v);
        }
    }
    ssum[tid] = s; smax[tid] = m;
    __syncthreads();
    for (int off = 128; off > 0; off >>= 1) {
        if (tid < off) {
            ssum[tid] += ssum[tid + off];
            smax[tid] = fmaxf(smax[tid], smax[tid + off]);
        }
        __syncthreads();
    }
    if (tid == 0) {
        if (c < HID) { pooled[c] = ssum[0] / (float)N_NODES; pooled[HID + c] = smax[0]; }
        else { const int cy = c - HID;
               pooled[256 + cy] = ssum[0] / (float)N_NODES; pooled[272 + cy] = smax[0]; }
    }
}

// ---------------------------------------------------------------------------
// Termination 2-layer MLP on pooled[288] -> scalar t. One block, 128 threads.
// ---------------------------------------------------------------------------
__global__ void __launch_bounds__(128)
term_mlp(const float* __restrict__ pooled,
         const float* __restrict__ Wt1, const float* __restrict__ bt1,
         const float* __restrict__ Wt2, const float* __restrict__ bt2,
         float* __restrict__ t_out)
{
    __shared__ float hred[128];
    const int j = threadIdx.x;
    float acc = bt1[j];
    for (int k = 0; k < 288; ++k)
        acc += pooled[k] * Wt1[(size_t)k * HID + j];
    acc = fmaxf(acc, 0.f);
    hred[j] = acc * Wt2[j];
    __syncthreads();
    for (int off = 64; off > 0; off >>= 1) {
        if (j < off) hred[j] += hred[j + off];
        __syncthreads();
    }
    if (j == 0) t_out[0] = hred[0] + bt2[0];
}

// ---------------------------------------------------------------------------
// Fill p with -inf using nontemporal stores (256 MB stream; keep L2 clean).
// ---------------------------------------------------------------------------
__global__ void __launch_bounds__(256)
fill_neg_inf(v4f* __restrict__ p, int n4)
{
    const v4f v = {-INFINITY, -INFINITY, -INFINITY, -INFINITY};
    for (int i = blockIdx.x * 256 + threadIdx.x; i < n4; i += 16384 * 256)
        __builtin_nontemporal_store(v, p + i);
}

// ---------------------------------------------------------------------------
// Predecessor edge scores scattered into dense p. One wave per edge:
// score = dot(cat(new_h[src], new_h[dst], w), Wp) + bp
// ---------------------------------------------------------------------------
__global__ void __launch_bounds__(256)
edge_pred(const float* __restrict__ newh,
          const int* __restrict__ src, const int* __restrict__ dst,
          const float* __restrict__ w,
          const float* __restrict__ Wp, const float* __restrict__ bp,
          float* __restrict__ p)
{
    const int e = blockIdx.x * 8 + (threadIdx.x >> 5);  // grid = E/8 exactly
    const int lane = threadIdx.x & 31;
    const int s = src[e], d = dst[e];
    const v4f hs = *(const v4f*)(newh + (size_t)s * HID + lane * 4);
    const v4f hd = *(const v4f*)(newh + (size_t)d * HID + lane * 4);
    const v4f w0 = *(const v4f*)(Wp + lane * 4);
    const v4f w1 = *(const v4f*)(Wp + HID + lane * 4);
    float part = hs[0]*w0[0] + hs[1]*w0[1] + hs[2]*w0[2] + hs[3]*w0[3]
               + hd[0]*w1[0] + hd[1]*w1[1] + hd[2]*w1[2] + hd[3]*w1[3];
#pragma unroll
    for (int off = 16; off > 0; off >>= 1)
        part += __shfl_xor(part, off, 32);
    if (lane == 0)
        p[(size_t)s * N_NODES + d] = part + w[e] * Wp[256] + bp[0];
}

// ---------------------------------------------------------------------------
// Host launcher
// ---------------------------------------------------------------------------
extern "C" void kernel_launch(void* const* d_in, const int* in_sizes, int n_in,
                              void* d_out, int out_size, void* d_ws, size_t ws_size,
                              hipStream_t stream)
{
    // Inputs in setup_inputs() order
    const float* x    = (const float*)d_in[0];   // [8192,16]
    const float* h    = (const float*)d_in[1];   // [8192,128]
    const int*   src  = (const int*)  d_in[2];   // [E]
    const int*   dst  = (const int*)  d_in[3];   // [E]
    const float* w    = (const float*)d_in[4];   // [E,1]
    const float* We   = (const float*)d_in[5];   // [144,128]
    const float* be   = (const float*)d_in[6];   // [128]
    const float* Wm   = (const float*)d_in[7];   // [257,128]
    const float* bm   = (const float*)d_in[8];   // [128]
    const float* Wu   = (const float*)d_in[9];   // [256,128]
    const float* bu   = (const float*)d_in[10];  // [128]
    const float* Wd   = (const float*)d_in[11];  // [256,16]
    const float* bd   = (const float*)d_in[12];  // [16]
    const float* Wt1  = (const float*)d_in[13];  // [288,128]
    const float* bt1  = (const float*)d_in[14];  // [128]
    const float* Wt2  = (const float*)d_in[15];  // [128,1]
    const float* bt2  = (const float*)d_in[16];  // [1]
    const float* Wp   = (const float*)d_in[17];  // [257,1]
    const float* bp   = (const float*)d_in[18];  // [1]

    // Output layout (flat, return order): y, p, new_h, t
    float* out = (float*)d_out;
    float* y_out    = out;                                   // 8192*16   = 131072
    float* p_out    = out + 131072;                          // 8192*8192 = 67108864
    float* newh_out = out + 131072 + 67108864;               // 8192*128  = 1048576
    float* t_out    = out + 131072 + 67108864 + 1048576;     // 1

    // Workspace carve-out (floats)
    float* ws     = (float*)d_ws;
    float* z      = ws;                    // 1048576
    float* Apre   = ws + 1048576;          // 1048576
    float* Bpre   = ws + 2 * 1048576;      // 1048576
    float* agg    = ws + 3 * 1048576;      // 1048576
    float* pooled = ws + 4 * 1048576;      // 288
    float* packs  = ws + 4 * 1048576 + 512;
    float* pkWe   = packs;                 // 36*8*64 = 18432
    float* pkWmT  = packs + 18432;         // 32*8*64 = 16384
    float* pkWmM  = packs + 35328;         // 16384 (padded offsets, all even)
    float* pkWu   = packs + 52224;         // 64*8*64 = 32768
    float* pkWd   = packs + 85504;         // 64*1*64 = 4096

    const dim3 blk256(256), blk128(128);

    // 0) Pack all weight matrices into WMMA B-fragment order
    pack_w<<<dim3(36),  blk256, 0, stream>>>(We,             pkWe,  HID,  8, 36); // K=144
    pack_w<<<dim3(64),  blk256, 0, stream>>>(Wm,             pkWmT, HID,  8, 32); // K=128
    pack_w<<<dim3(64),  blk256, 0, stream>>>(Wm + 128 * HID, pkWmM, HID,  8, 32); // K=128
    pack_w<<<dim3(128), blk256, 0, stream>>>(Wu,             pkWu,  HID,  8, 64); // K=256
    pack_w<<<dim3(8),   blk256, 0, stream>>>(Wd,             pkWd,  OUTD, 1, 64); // K=256

    // 1) Encoder: z = cat(x,h) @ We + be      (M=8192, K=16+128, N=128)
    wmma_linear<<<dim3(512), blk256, 0, stream>>>(x, 16, h, HID, pkWe, be, z, HID, 8);

    // 2) A = z @ Wm[0:128]   3) B = z @ Wm[128:256] + bm
    wmma_linear<<<dim3(512), blk256, 0, stream>>>(z, HID, nullptr, 0, pkWmT, nullptr, Apre, HID, 8);
    wmma_linear<<<dim3(512), blk256, 0, stream>>>(z, HID, nullptr, 0, pkWmM, bm,      Bpre, HID, 8);

    // 4) agg = -inf ; 5) per-edge message + atomic segment-max ; 6) +-inf -> 0
    init_agg<<<dim3(1024), blk256, 0, stream>>>((v4u*)agg);
    edge_msg_max<<<dim3(N_EDGES / 8), blk256, 0, stream>>>(Apre, Bpre, Wm + 256 * HID,
                                                           src, dst, w, agg);
    fixup_agg<<<dim3(4096), blk256, 0, stream>>>(agg);

    // 7) new_h = cat(z, agg) @ Wu + bu  -> straight into d_out slot
    wmma_linear<<<dim3(512), blk256, 0, stream>>>(z, HID, agg, HID, pkWu, bu, newh_out, HID, 8);

    // 8) y = cat(z, new_h) @ Wd + bd    (N=16)
    wmma_linear<<<dim3(64), blk256, 0, stream>>>(z, HID, newh_out, HID, pkWd, bd, y_out, OUTD, 1);

    // 9) pooled stats ; 10) termination MLP -> t
    col_reduce<<<dim3(144), blk256, 0, stream>>>(newh_out, y_out, pooled);
    term_mlp<<<dim3(1), blk128, 0, stream>>>(pooled, Wt1, bt1, Wt2, bt2, t_out);

    // 11) p = -inf (nontemporal streaming fill, 16M float4)
    fill_neg_inf<<<dim3(16384), blk256, 0, stream>>>((v4f*)p_out, 67108864 / 4);

    // 12) scatter predecessor edge scores into p
    edge_pred<<<dim3(N_EDGES / 8), blk256, 0, stream>>>(newh_out, src, dst, w, Wp, bp, p_out);
}